// Actor_47021301956935
// MI455X (gfx1250) — compile-verified
//
#include <hip/hip_runtime.h>
#include <hip/hip_bf16.h>

// ---------------------------------------------------------------------------
// Problem constants (from the reference)
// ---------------------------------------------------------------------------
#define NNODES   50000
#define NREG     8
#define DIM      128
#define GDIM     147
#define ATOT     100000
#define HID      256
// IN_DIM = 2*128 + 8*128 + 147 = 1427 ; W1 rows: [0:128)=node, [128:256)=region,
// [256:1280)=flattened region table, [1280:1427)=global context.

#define STRIDE   260            // LDS row stride (floats); 260%64==4 -> conflict-free
#define MTILE    64             // actions per workgroup in the fused MLP

typedef float v2f __attribute__((ext_vector_type(2)));
typedef float v8f __attribute__((ext_vector_type(8)));

#ifndef __has_builtin
#define __has_builtin(x) 0
#endif
#if __has_builtin(__builtin_amdgcn_wmma_f32_16x16x4_f32)
#define HAVE_WMMA_F32 1
#else
#define HAVE_WMMA_F32 0
#endif

// D(16x16,f32) = A(16x4,f32) x B(4x16,f32) + C   — CDNA5 V_WMMA_F32_16X16X4_F32
// A layout: lane<16: VGPR0=A[lane][4k+0], VGPR1=A[lane][4k+1]
//           lane>=16: VGPR0=A[lane-16][4k+2], VGPR1=A[lane-16][4k+3]
// B layout mirrors A (K rows striped across lane halves); C/D: VGPR v holds
// row v (lanes 0-15) and row v+8 (lanes 16-31), N = lane&15.
__device__ __forceinline__ v8f wmma4(v2f a, v2f b, v8f c) {
#if HAVE_WMMA_F32
  return __builtin_amdgcn_wmma_f32_16x16x4_f32(
      /*neg_a=*/false, a, /*neg_b=*/false, b,
      /*c_mod=*/(short)0, c, /*reuse_a=*/false, /*reuse_b=*/false);
#else
  // Cross-lane emulation (correct, used only if the builtin is unavailable).
  const int lane = threadIdx.x & 31;
  const int n    = lane & 15;
  const int half = lane >> 4;
  float b0 = __shfl(b.x, n,      32);
  float b1 = __shfl(b.y, n,      32);
  float b2 = __shfl(b.x, n + 16, 32);
  float b3 = __shfl(b.y, n + 16, 32);
#pragma unroll
  for (int v = 0; v < 8; ++v) {
    int m = v + 8 * half;
    float a0 = __shfl(a.x, m,      32);
    float a1 = __shfl(a.y, m,      32);
    float a2 = __shfl(a.x, m + 16, 32);
    float a3 = __shfl(a.y, m + 16, 32);
    c[v] += a0 * b0 + a1 * b1 + a2 * b2 + a3 * b3;
  }
  return c;
#endif
}

// ---------------------------------------------------------------------------
// Kernel 0: LDS-tiled transpose  dst[c][r] = src[r][c]   (one-time weight prep)
// block (16,16); grid (cols/16, rows/16). All dims here are multiples of 16.
// ---------------------------------------------------------------------------
__global__ __launch_bounds__(256) void transpose_mat(
    const float* __restrict__ src, float* __restrict__ dst, int rows,
    int cols) {
  __shared__ float tile[16][17];
  const int bx = blockIdx.x * 16;   // column offset in src
  const int by = blockIdx.y * 16;   // row offset in src
  const int tx = threadIdx.x, ty = threadIdx.y;
  tile[ty][tx] = src[(long)(by + ty) * cols + (bx + tx)];
  __syncthreads();
  // dst dims are [cols][rows]
  dst[(long)(bx + ty) * rows + (by + tx)] = tile[tx][ty];
}

// ---------------------------------------------------------------------------
// Kernel 1: nodeW1[n][h] = node_embeddings[n][:] @ W1[0:128][h]   (WMMA GEMM)
// Uses transposed weights W1t[h][k] so B fragments are single b64 loads.
// grid.x = NNODES/16 = 3125, block = 256 (8 waves; wave w owns cols w*32..+31)
// ---------------------------------------------------------------------------
__global__ __launch_bounds__(256) void node_precompute(
    const float* __restrict__ emb, const float* __restrict__ W1t,
    float* __restrict__ nodeW1) {
  const int tid  = threadIdx.x;
  const int lane = tid & 31, wave = tid >> 5;
  const int half = lane >> 4, l15 = lane & 15;
  const int m0 = blockIdx.x * 16;
  const int n0 = wave * 32;

  v8f c0 = {}, c1 = {};
  const float* arow = emb + (long)(m0 + l15) * DIM + 2 * half;
  const float* w0   = W1t + (long)(n0 + l15) * DIM + 2 * half;
  const float* w1   = W1t + (long)(n0 + 16 + l15) * DIM + 2 * half;
#pragma unroll 4
  for (int k = 0; k < 32; ++k) {
    v2f a, b;
    a.x = arow[4 * k];
    a.y = arow[4 * k + 1];
    b.x = w0[4 * k];
    b.y = w0[4 * k + 1];
    c0 = wmma4(a, b, c0);
    b.x = w1[4 * k];
    b.y = w1[4 * k + 1];
    c1 = wmma4(a, b, c1);
  }
#pragma unroll
  for (int v = 0; v < 8; ++v) {
    int row = m0 + v + 8 * half;
    nodeW1[(long)row * HID + n0 + l15]      = c0[v];
    nodeW1[(long)row * HID + n0 + 16 + l15] = c1[v];
  }
}

// ---------------------------------------------------------------------------
// Kernel 2: regW1[r][h] = region_emb[r] @ W1[128:256][h]
//           tailW1[h]   = b1[h] + tail @ W1[256:1427][h]
// one block of 256 threads, thread = output column h
// ---------------------------------------------------------------------------
__global__ __launch_bounds__(256) void tail_precompute(
    const float* __restrict__ re, const float* __restrict__ gc,
    const float* __restrict__ W1, const float* __restrict__ b1,
    float* __restrict__ regW1, float* __restrict__ tailW1) {
  const int c = threadIdx.x;
  for (int r = 0; r < NREG; ++r) {
    float s = 0.f;
    for (int k = 0; k < DIM; ++k)
      s += re[r * DIM + k] * W1[(long)(DIM + k) * HID + c];
    regW1[r * HID + c] = s;
  }
  float s = b1[c];
  for (int k = 0; k < NREG * DIM; ++k)           // rows 256..1279
    s += re[k] * W1[(long)(2 * DIM + k) * HID + c];
  for (int k = 0; k < GDIM; ++k)                 // rows 1280..1426
    s += gc[k] * W1[(long)(2 * DIM + NREG * DIM + k) * HID + c];
  tailW1[c] = s;
}

// ---------------------------------------------------------------------------
// Fused MLP layer: dst[64][256] = relu(src[64][256] @ W + bias), W given
// TRANSPOSED as Wt[col][k] so each B fragment is one aligned b64 load.
// 8 waves: wave w -> rows (w>>1)*16..+15, cols (w&1)*128..+127 (8 N-tiles)
// ---------------------------------------------------------------------------
__device__ __forceinline__ void layer_gemm(
    const float* __restrict__ src,   // LDS [MTILE][STRIDE]
    const float* __restrict__ Wt,    // global [256][256], transposed
    const float* __restrict__ bias,  // global [256]
    float* __restrict__ dst,         // LDS [MTILE][STRIDE]
    int wave, int lane) {
  const int m0    = (wave >> 1) * 16;
  const int nbase = (wave & 1) * 128;
  const int half  = lane >> 4, l15 = lane & 15;

  v8f acc[8];
#pragma unroll
  for (int t = 0; t < 8; ++t) acc[t] = (v8f){};

  const float* arow = src + (m0 + l15) * STRIDE + 2 * half;
  const float* wrow[8];
#pragma unroll
  for (int t = 0; t < 8; ++t)
    wrow[t] = Wt + (long)(nbase + t * 16 + l15) * HID + 2 * half;

  for (int k = 0; k < 64; ++k) {
    v2f a;
    a.x = arow[4 * k];
    a.y = arow[4 * k + 1];
#pragma unroll
    for (int t = 0; t < 8; ++t) {
      v2f b;
      b.x = wrow[t][4 * k];
      b.y = wrow[t][4 * k + 1];
      acc[t] = wmma4(a, b, acc[t]);
    }
  }
#pragma unroll
  for (int t = 0; t < 8; ++t) {
    const int col = nbase + t * 16 + l15;
    const float bb = bias[col];
#pragma unroll
    for (int v = 0; v < 8; ++v) {
      const int row = m0 + v + 8 * half;
      dst[row * STRIDE + col] = fmaxf(acc[t][v] + bb, 0.f);
    }
  }
}

// ---------------------------------------------------------------------------
// Kernel 3: fused per-action MLP. 64 actions / block, 256 threads.
// logits -> out[ATOT + a]
// ---------------------------------------------------------------------------
__global__ __launch_bounds__(256) void mlp_fused(
    const float* __restrict__ nodeW1, const float* __restrict__ regW1,
    const float* __restrict__ tailW1,
    const float* __restrict__ W2t, const float* __restrict__ b2,
    const float* __restrict__ W3t, const float* __restrict__ b3,
    const float* __restrict__ W4, const float* __restrict__ b4,
    const int* __restrict__ act_nodes, const int* __restrict__ act_regions,
    float* __restrict__ out) {
  extern __shared__ float lds[];
  float* hA = lds;                     // [MTILE][STRIDE]
  float* hB = lds + MTILE * STRIDE;    // [MTILE][STRIDE]

  const int tid  = threadIdx.x;
  const int lane = tid & 31, wave = tid >> 5;
  const int a0 = blockIdx.x * MTILE;

  // ---- layer 1: gather precomputed pieces, add, ReLU -> hA
  for (int it = 0; it < MTILE; ++it) {
    int a  = a0 + it;
    int aa = (a < ATOT) ? a : 0;            // clamp dummy tail rows
    int node = act_nodes[aa];
    int reg  = act_regions[aa];
    float v = nodeW1[(long)node * HID + tid] + regW1[reg * HID + tid] +
              tailW1[tid];
    hA[it * STRIDE + tid] = fmaxf(v, 0.f);
  }
  __syncthreads();

  // ---- layer 2: hB = relu(hA @ W2 + b2)
  layer_gemm(hA, W2t, b2, hB, wave, lane);
  __syncthreads();

  // ---- layer 3: hA = relu(hB @ W3 + b3)
  layer_gemm(hB, W3t, b3, hA, wave, lane);
  __syncthreads();

  // ---- layer 4: logit = hA @ W4 + b4  (one thread per action row)
  if (tid < MTILE) {
    const float* h = hA + tid * STRIDE;
    float s = 0.f;
#pragma unroll 8
    for (int k = 0; k < HID; ++k) s += h[k] * W4[k];
    s += b4[0];
    int a = a0 + tid;
    if (a < ATOT) out[ATOT + a] = s;
  }
}

// ---------------------------------------------------------------------------
// Kernel 4: softmax stats (max, sum of exp) over logits; single block.
// ---------------------------------------------------------------------------
__global__ __launch_bounds__(1024) void softmax_stats(
    const float* __restrict__ logits, float* __restrict__ red, int n) {
  __shared__ float  sm[1024];
  __shared__ double sd[1024];
  const int tid = threadIdx.x;

  float m = -INFINITY;
  for (int i = tid; i < n; i += 1024) m = fmaxf(m, logits[i]);
  sm[tid] = m;
  __syncthreads();
  for (int s = 512; s > 0; s >>= 1) {
    if (tid < s) sm[tid] = fmaxf(sm[tid], sm[tid + s]);
    __syncthreads();
  }
  const float gmax = sm[0];
  __syncthreads();

  double acc = 0.0;
  for (int i = tid; i < n; i += 1024) acc += (double)expf(logits[i] - gmax);
  sd[tid] = acc;
  __syncthreads();
  for (int s = 512; s > 0; s >>= 1) {
    if (tid < s) sd[tid] += sd[tid + s];
    __syncthreads();
  }
  if (tid == 0) {
    red[0] = gmax;
    red[1] = (float)sd[0];
  }
}

// ---------------------------------------------------------------------------
// Kernel 5: probs[a] = exp(logit - max) / sum
// ---------------------------------------------------------------------------
__global__ __launch_bounds__(256) void softmax_write(
    const float* __restrict__ logits, const float* __restrict__ red,
    float* __restrict__ probs, int n) {
  int a = blockIdx.x * blockDim.x + threadIdx.x;
  if (a < n) {
    float inv = 1.0f / red[1];
    probs[a] = expf(logits[a] - red[0]) * inv;
  }
}

// ---------------------------------------------------------------------------
// Launch
// ---------------------------------------------------------------------------
extern "C" void kernel_launch(void* const* d_in, const int* in_sizes, int n_in,
                              void* d_out, int out_size, void* d_ws,
                              size_t ws_size, hipStream_t stream) {
  const float* node_emb = (const float*)d_in[0];
  const float* reg_emb  = (const float*)d_in[1];
  const float* gctx     = (const float*)d_in[2];
  const float* W1       = (const float*)d_in[3];
  const float* b1       = (const float*)d_in[4];
  const float* W2       = (const float*)d_in[5];
  const float* b2       = (const float*)d_in[6];
  const float* W3       = (const float*)d_in[7];
  const float* b3       = (const float*)d_in[8];
  const float* W4       = (const float*)d_in[9];
  const float* b4       = (const float*)d_in[10];
  const int* act_nodes   = (const int*)d_in[11];
  const int* act_regions = (const int*)d_in[12];
  float* out = (float*)d_out;   // [0:ATOT)=probs, [ATOT:2*ATOT)=logits

  float* ws      = (float*)d_ws;
  float* nodeW1  = ws;                                  // 50000*256 floats
  float* regW1   = nodeW1 + (long)NNODES * HID;         // 8*256
  float* tailW1  = regW1 + NREG * HID;                  // 256
  float* red     = tailW1 + HID;                        // 2 (pad to 16)
  float* W1t     = red + 16;                            // 256*128
  float* W2t     = W1t + HID * DIM;                     // 256*256
  float* W3t     = W2t + HID * HID;                     // 256*256

  // One-time weight transposes (Wt[col][k]) for b64 B-fragment loads.
  dim3 tb(16, 16);
  transpose_mat<<<dim3(HID / 16, DIM / 16), tb, 0, stream>>>(W1, W1t, DIM, HID);
  transpose_mat<<<dim3(HID / 16, HID / 16), tb, 0, stream>>>(W2, W2t, HID, HID);
  transpose_mat<<<dim3(HID / 16, HID / 16), tb, 0, stream>>>(W3, W3t, HID, HID);

  node_precompute<<<NNODES / 16, 256, 0, stream>>>(node_emb, W1t, nodeW1);
  tail_precompute<<<1, 256, 0, stream>>>(reg_emb, gctx, W1, b1, regW1, tailW1);

  const int nblk = (ATOT + MTILE - 1) / MTILE;          // 1563
  const size_t lds_bytes = 2u * MTILE * STRIDE * sizeof(float);  // 130 KB
  mlp_fused<<<nblk, 256, lds_bytes, stream>>>(nodeW1, regW1, tailW1, W2t, b2,
                                              W3t, b3, W4, b4, act_nodes,
                                              act_regions, out);

  softmax_stats<<<1, 1024, 0, stream>>>(out + ATOT, red, ATOT);
  softmax_write<<<(ATOT + 255) / 256, 256, 0, stream>>>(out + ATOT, red, out,
                                                        ATOT);
}